// GNNEncoder2_90744069030632
// MI455X (gfx1250) — compile-verified
//
#include <hip/hip_runtime.h>

#define DIM 128
#define NEG_SLOPE 0.2f

typedef float v2f __attribute__((ext_vector_type(2)));
typedef float v8f __attribute__((ext_vector_type(8)));

// Order-preserving float <-> uint mapping for atomicMax-based segment max.
// enc(-inf)=0x007FFFFF > 0, so 0 is a safe "no edges" sentinel.
__device__ __forceinline__ unsigned enc_f(float f) {
    unsigned u = __float_as_uint(f);
    return (u & 0x80000000u) ? ~u : (u | 0x80000000u);
}
__device__ __forceinline__ float dec_f(unsigned u) {
    unsigned v = (u & 0x80000000u) ? (u & 0x7FFFFFFFu) : ~u;
    return __uint_as_float(v);
}

// ---------------------------------------------------------------------------
// Node transform: x_l = x@W_l + b_l, x_r = x@W_r + b_r
//   - persistent blocks: each wave keeps ALL its B fragments (32 x v2f = 64
//     VGPRs) in registers, so the tile loop is pure {async copy, LDS, WMMA}.
//   - x tile (16x128 f32 = 8 KB) staged in LDS via GLOBAL_LOAD_ASYNC_TO_LDS_B128
//     (ASYNCcnt), consumed by V_WMMA_F32_16X16X4_F32 (exact fp32 MMA).
// Block = 512 threads = 16 wave32s:
//   waves 0..7  -> x_l col-tiles 0..7 ; waves 8..15 -> x_r col-tiles 0..7
// ---------------------------------------------------------------------------
__global__ __launch_bounds__(512) void gat_transform(
    const float* __restrict__ x,  const float* __restrict__ Wl,
    const float* __restrict__ bl, const float* __restrict__ Wr,
    const float* __restrict__ br, float* __restrict__ xl,
    float* __restrict__ xr, int n_nodes, int n_tiles)
{
    __shared__ float xt[16][DIM + 4];   // pad: rows land on distinct LDS banks

    const int tid     = threadIdx.x;
    const int wave    = tid >> 5;
    const int lane    = tid & 31;
    const int mat     = wave >> 3;       // 0 -> W_l path, 1 -> W_r path
    const int colTile = wave & 7;

    const float* __restrict__ W = mat ? Wr : Wl;
    const float* __restrict__ b = mat ? br : bl;
    float*       __restrict__ o = mat ? xr : xl;

    const int half = lane >> 4;          // 0: lanes 0-15, 1: lanes 16-31
    const int mrow = lane & 15;          // A-row (and B/D column) index
    const int col  = colTile * 16 + mrow;

    // Preload this wave's B fragments for the whole K=128 into registers.
    // ISA 16x4 f32 fragment layout: lanes 0-15 hold K={k,k+1}, 16-31 K={k+2,k+3}.
    v2f bfr[32];
    #pragma unroll
    for (int j = 0; j < 32; ++j) {
        const int kk = 4 * j + 2 * half;
        bfr[j].x = W[(size_t)kk * DIM + col];
        bfr[j].y = W[(size_t)(kk + 1) * DIM + col];
    }
    const float bias = b[col];

    // Async-copy indexing: thread t moves one 16B chunk of the 16x128 tile.
    const int lrow   = tid >> 5;         // 0..15
    const int lchunk = tid & 31;         // 0..31 chunks of 4 floats per row
    const unsigned lds_off =
        (unsigned)(uintptr_t)(&xt[lrow][lchunk * 4]);

    for (int tile = blockIdx.x; tile < n_tiles; tile += gridDim.x) {
        const int node0 = tile * 16;
        const bool full = (node0 + 16 <= n_nodes);

        if (full) {
            const float* gsrc = x + (size_t)(node0 + lrow) * DIM + lchunk * 4;
            asm volatile("global_load_async_to_lds_b128 %0, %1, off"
                         :: "v"(lds_off), "v"(gsrc) : "memory");
        } else {
            for (int i = tid; i < 16 * DIM; i += 512) {
                const int r = i / DIM, c = i % DIM;
                const int node = node0 + r;
                xt[r][c] = (node < n_nodes) ? x[(size_t)node * DIM + c] : 0.0f;
            }
        }
        asm volatile("s_wait_asynccnt 0x0" ::: "memory");
        __syncthreads();

        v8f acc = {};
        #pragma unroll
        for (int j = 0; j < 32; ++j) {
            const int kk = 4 * j + 2 * half;
            v2f a;
            a.x = xt[mrow][kk];
            a.y = xt[mrow][kk + 1];
            acc = __builtin_amdgcn_wmma_f32_16x16x4_f32(
                /*neg_a=*/false, a, /*neg_b=*/false, bfr[j],
                /*c_mod=*/(short)0, acc, /*reuse_a=*/false, /*reuse_b=*/false);
        }

        // C/D layout: VGPR r holds rows M=r (lanes 0-15) / M=r+8 (lanes 16-31)
        if (full) {
            #pragma unroll
            for (int r = 0; r < 8; ++r)
                o[(size_t)(node0 + r + 8 * half) * DIM + col] = acc[r] + bias;
        } else {
            #pragma unroll
            for (int r = 0; r < 8; ++r) {
                const int node = node0 + r + 8 * half;
                if (node < n_nodes)
                    o[(size_t)node * DIM + col] = acc[r] + bias;
            }
        }
        __syncthreads();   // protect LDS tile before next iteration's copy
    }
}

// ---------------------------------------------------------------------------
// Per-edge attention logits + segment max (atomicMax on monotone encoding).
// One wave32 per edge; lane l owns float4 #l of the 128-dim feature row.
// ---------------------------------------------------------------------------
__global__ __launch_bounds__(256) void gat_edge_logits(
    const float* __restrict__ xl, const float* __restrict__ xr,
    const int* __restrict__ ei, float* __restrict__ e_out,
    unsigned* __restrict__ e_max_u, const float* __restrict__ att,
    int n_edges)
{
    const int lane = threadIdx.x & 31;
    const int edge = (int)((blockIdx.x * blockDim.x + threadIdx.x) >> 5);
    if (edge >= n_edges) return;

    const int src = ei[edge];
    const int dst = ei[n_edges + edge];

    const float4 a4 = ((const float4*)att)[lane];
    const float4 l4 = ((const float4*)(xl + (size_t)src * DIM))[lane];
    const float4 r4 = ((const float4*)(xr + (size_t)dst * DIM))[lane];

    float s = 0.f, h;
    h = l4.x + r4.x; h = (h >= 0.f) ? h : NEG_SLOPE * h; s += h * a4.x;
    h = l4.y + r4.y; h = (h >= 0.f) ? h : NEG_SLOPE * h; s += h * a4.y;
    h = l4.z + r4.z; h = (h >= 0.f) ? h : NEG_SLOPE * h; s += h * a4.z;
    h = l4.w + r4.w; h = (h >= 0.f) ? h : NEG_SLOPE * h; s += h * a4.w;

    #pragma unroll
    for (int off = 16; off > 0; off >>= 1)
        s += __shfl_xor(s, off, 32);

    if (lane == 0) {
        e_out[edge] = s;
        atomicMax(&e_max_u[dst], enc_f(s));
    }
}

// ---------------------------------------------------------------------------
// alpha = exp(e - max[dst]); denom[dst] += alpha
// ---------------------------------------------------------------------------
__global__ __launch_bounds__(256) void gat_alpha(
    const int* __restrict__ ei, float* __restrict__ e_buf,
    const unsigned* __restrict__ e_max_u, float* __restrict__ denom,
    int n_edges)
{
    const int edge = blockIdx.x * blockDim.x + threadIdx.x;
    if (edge >= n_edges) return;
    const int dst = ei[n_edges + edge];
    const unsigned mu = e_max_u[dst];
    const float m = (mu == 0u) ? 0.0f : dec_f(mu);   // matches isfinite->0 fixup
    const float a = __expf(e_buf[edge] - m);
    e_buf[edge] = a;
    atomicAdd(&denom[dst], a);
}

// ---------------------------------------------------------------------------
// out[dst] += (alpha/denom[dst]) * x_l[src]; one wave per edge, float atomics.
// ---------------------------------------------------------------------------
__global__ __launch_bounds__(256) void gat_aggregate(
    const float* __restrict__ xl, const int* __restrict__ ei,
    const float* __restrict__ alpha, const float* __restrict__ denom,
    float* __restrict__ out, int n_edges)
{
    const int lane = threadIdx.x & 31;
    const int edge = (int)((blockIdx.x * blockDim.x + threadIdx.x) >> 5);
    if (edge >= n_edges) return;

    const int src = ei[edge];
    const int dst = ei[n_edges + edge];
    const float w = alpha[edge] / (denom[dst] + 1e-16f);

    const float4 l4 = ((const float4*)(xl + (size_t)src * DIM))[lane];
    float* o = out + (size_t)dst * DIM + lane * 4;
    atomicAdd(o + 0, w * l4.x);
    atomicAdd(o + 1, w * l4.y);
    atomicAdd(o + 2, w * l4.z);
    atomicAdd(o + 3, w * l4.w);
}

__global__ void gat_finish(float* __restrict__ out,
                           const float* __restrict__ bias, int total)
{
    const int i = blockIdx.x * blockDim.x + threadIdx.x;
    if (i < total) {
        const float v = out[i] + bias[i & (DIM - 1)];
        out[i] = (v > 0.f) ? v : 0.f;
    }
}

__global__ void gat_init(float* __restrict__ out, float* __restrict__ denom,
                         unsigned* __restrict__ e_max_u, int n_nodes)
{
    const int i = blockIdx.x * blockDim.x + threadIdx.x;
    if (i < n_nodes * DIM) out[i] = 0.f;
    if (i < n_nodes) { denom[i] = 0.f; e_max_u[i] = 0u; }
}

// ---------------------------------------------------------------------------
extern "C" void kernel_launch(void* const* d_in, const int* in_sizes, int n_in,
                              void* d_out, int out_size, void* d_ws, size_t ws_size,
                              hipStream_t stream)
{
    const float* x    = (const float*)d_in[0];
    const int*   ei   = (const int*)  d_in[1];   // (2, E): [0..E)=src, [E..2E)=dst
    const float* Wl   = (const float*)d_in[2];
    const float* bl   = (const float*)d_in[3];
    const float* Wr   = (const float*)d_in[4];
    const float* br   = (const float*)d_in[5];
    const float* att  = (const float*)d_in[6];
    const float* bias = (const float*)d_in[7];
    float* out = (float*)d_out;

    const int n_nodes = in_sizes[0] / DIM;
    const int n_edges = in_sizes[1] / 2;

    // Carve workspace (256B-aligned sections).
    char* ws = (char*)d_ws;
    size_t off = 0;
    auto carve = [&](size_t bytes) -> char* {
        char* p = ws + off;
        off += (bytes + 255) & ~(size_t)255;
        return p;
    };
    float*    xl    = (float*)   carve((size_t)n_nodes * DIM * sizeof(float));
    float*    xr    = (float*)   carve((size_t)n_nodes * DIM * sizeof(float));
    float*    ebuf  = (float*)   carve((size_t)n_edges * sizeof(float));
    unsigned* emax  = (unsigned*)carve((size_t)n_nodes * sizeof(unsigned));
    float*    denom = (float*)   carve((size_t)n_nodes * sizeof(float));

    const int total  = n_nodes * DIM;
    const int n_tiles = (n_nodes + 15) / 16;
    const int persistentBlocks = n_tiles < 2048 ? n_tiles : 2048;
    const int edgeWaveBlocks = (int)(((size_t)n_edges * 32 + 255) / 256);

    gat_init<<<(total + 255) / 256, 256, 0, stream>>>(out, denom, emax, n_nodes);
    gat_transform<<<persistentBlocks, 512, 0, stream>>>(
        x, Wl, bl, Wr, br, xl, xr, n_nodes, n_tiles);
    gat_edge_logits<<<edgeWaveBlocks, 256, 0, stream>>>(
        xl, xr, ei, ebuf, emax, att, n_edges);
    gat_alpha<<<(n_edges + 255) / 256, 256, 0, stream>>>(
        ei, ebuf, emax, denom, n_edges);
    gat_aggregate<<<edgeWaveBlocks, 256, 0, stream>>>(
        xl, ei, ebuf, denom, out, n_edges);
    gat_finish<<<(total + 255) / 256, 256, 0, stream>>>(out, bias, total);
}